// SUCAM_78486232367649
// MI455X (gfx1250) — compile-verified
//
#include <hip/hip_runtime.h>

typedef __attribute__((ext_vector_type(2))) float v2f;
typedef __attribute__((ext_vector_type(8))) float v8f;

#define N_CAM     6
#define BN_COUNT  12            // B*N
#define D_COUNT   112
#define HF        28
#define WF        60
#define HW        (HF * WF)     // 1680
#define DHW       (D_COUNT * HW)// 188160 (divisible by 16)
#define TOTAL_PTS (2 * N_CAM * DHW) // 2257920
#define C_FEAT    64
#define BEV_X     200
#define BEV_Y     200
#define XY        (BEV_X * BEV_Y)   // 40000
#define OUT_PER_B (C_FEAT * XY)     // 2560000

// ---------------- zero output (harness poisons d_out) ----------------
__global__ void zero_kernel(float4* __restrict__ out, int n4) {
  int i = blockIdx.x * blockDim.x + threadIdx.x;
  if (i < n4) out[i] = float4{0.f, 0.f, 0.f, 0.f};
}

// ------- prep: comb = R @ inv(K), pre-swizzled into WMMA A layout -------
// ws layout: per (b,n): 64 floats = per-lane (a.x, a.y) pairs for the
// 16x4 f32 A matrix (lane<16: K=0,1 ; lane>=16: K=2,3 ; rows>=3 are zero).
__global__ void prep_kernel(const float* __restrict__ intr,
                            const float* __restrict__ ext,
                            float* __restrict__ ws) {
  int t = threadIdx.x;
  if (t >= BN_COUNT) return;
  const float* K = intr + t * 9;
  const float* E = ext + t * 16;
  float a = K[0], b = K[1], c = K[2];
  float d = K[3], e = K[4], f = K[5];
  float g = K[6], h = K[7], i = K[8];
  float A  =  (e * i - f * h);
  float B  = -(d * i - f * g);
  float Cc =  (d * h - e * g);
  float s  = 1.0f / (a * A + b * B + c * Cc);
  float inv[9];
  inv[0] = A * s;  inv[1] = (c * h - b * i) * s;  inv[2] = (b * f - c * e) * s;
  inv[3] = B * s;  inv[4] = (a * i - c * g) * s;  inv[5] = (c * d - a * f) * s;
  inv[6] = Cc * s; inv[7] = (b * g - a * h) * s;  inv[8] = (a * e - b * d) * s;
  float comb[9];
  for (int r = 0; r < 3; ++r)
    for (int cc = 0; cc < 3; ++cc)
      comb[r * 3 + cc] = E[r * 4 + 0] * inv[0 * 3 + cc] +
                         E[r * 4 + 1] * inv[1 * 3 + cc] +
                         E[r * 4 + 2] * inv[2 * 3 + cc];
  float trans[3] = { E[3], E[7], E[11] };
  float* o = ws + t * 64;
  for (int L = 0; L < 32; ++L) {
    int r = L & 15;
    float v0 = 0.f, v1 = 0.f;
    if (r < 3) {
      if (L < 16) { v0 = comb[r * 3 + 0]; v1 = comb[r * 3 + 1]; }
      else        { v0 = comb[r * 3 + 2]; v1 = trans[r]; }
    }
    o[L * 2 + 0] = v0;
    o[L * 2 + 1] = v1;
  }
}

// ---------------- main: WMMA geometry + atomic BEV splat ----------------
__global__ void __launch_bounds__(256)
bev_splat_kernel(const float* __restrict__ feats,
                 const float* __restrict__ combA,
                 float* __restrict__ out) {
  const int lane = threadIdx.x & 31;
  const int wid  = (blockIdx.x * blockDim.x + threadIdx.x) >> 5;
  const int base_p = wid * 16;        // 16 consecutive frustum points / wave
  const int bn = base_p / DHW;        // uniform across the group of 16
  const int b  = bn / N_CAM;

  // Per-lane point decode (lanes 16..31 mirror lanes 0..15: they carry
  // the K=2,3 rows of the B matrix for the same 16 points).
  const int r   = lane & 15;
  int local = base_p + r - bn * DHW;
  int di  = local / HW;
  int rem = local - di * HW;
  int h   = rem / WF;
  int w   = rem - h * WF;
  float dd = 2.0f + 0.5f * (float)di;            // depth bin value
  float xs = (float)w * (479.0f / 59.0f);        // linspace(0, 479, 60)
  float ys = (float)h * (223.0f / 27.0f);        // linspace(0, 223, 28)

  // B matrix (4x16): homogeneous point columns [x*d, y*d, d, 1]
  v2f bmat;
  bmat.x = (lane < 16) ? xs * dd : dd;    // K=0 : K=2
  bmat.y = (lane < 16) ? ys * dd : 1.0f;  // K=1 : K=3
  // A matrix (16x4): [comb | trans], pre-swizzled per lane
  v2f amat = ((const v2f*)(combA + bn * 64))[lane];

  v8f acc = {};
  // D = A x B : lane n<16 gets geom components 0..2 of point n in acc[0..2]
  acc = __builtin_amdgcn_wmma_f32_16x16x4_f32(false, amat, false, bmat,
                                              (short)0, acc, false, false);

  // Voxelize (valid in lanes 0..15; other lanes' enc is never read)
  float gx = acc[0], gy = acc[1], gz = acc[2];
  float fx = (gx + 50.0f) / 0.5f;   // (geom - (start - res/2)) / res
  float fy = (gy + 50.0f) / 0.5f;
  float fz = (gz + 10.0f) / 20.0f;
  int vx = (int)fx;                 // truncation == .astype(int32)
  int vy = (int)fy;
  int vz = (int)fz;
  bool kept = (vx >= 0) && (vx < BEV_X) &&
              (vy >= 0) && (vy < BEV_Y) &&
              (vz >= 0) && (vz < 1);
  int enc = kept ? (b * OUT_PER_B + vx * BEV_Y + vy) : -1;

  // Scatter: per kept point, coalesced 256B feature row + 64 f32 atomics.
  // readlane -> enc lands in an SGPR: reject test is a scalar branch and
  // all addresses are uniform-base + lane-offset.
  const float* frow = feats + (size_t)base_p * C_FEAT;
#pragma unroll 1
  for (int j = 0; j < 16; ++j) {
    int e = __builtin_amdgcn_readlane(enc, j);
    if (e < 0) continue;            // z-gating rejects ~88% of points
    const float* fv = frow + j * C_FEAT;
    // streamed once: non-temporal so L2 stays reserved for the atomic target
    float f0 = __builtin_nontemporal_load(fv + lane);
    float f1 = __builtin_nontemporal_load(fv + lane + 32);
    unsafeAtomicAdd(out + e + lane * XY, f0);
    unsafeAtomicAdd(out + e + (lane + 32) * XY, f1);
  }
}

extern "C" void kernel_launch(void* const* d_in, const int* in_sizes, int n_in,
                              void* d_out, int out_size, void* d_ws, size_t ws_size,
                              hipStream_t stream) {
  const float* feats = (const float*)d_in[0];   // (2,6,112,28,60,64) f32
  const float* intr  = (const float*)d_in[1];   // (2,6,3,3) f32
  const float* ext   = (const float*)d_in[2];   // (2,6,4,4) f32
  float* out = (float*)d_out;                   // (2, 64, 200, 200) f32
  float* ws  = (float*)d_ws;                    // 12*64 floats used

  int n4 = out_size / 4;
  zero_kernel<<<(n4 + 255) / 256, 256, 0, stream>>>((float4*)out, n4);
  prep_kernel<<<1, 32, 0, stream>>>(intr, ext, ws);
  // 2,257,920 pts / 16 per wave = 141,120 waves / 8 waves per block
  bev_splat_kernel<<<TOTAL_PTS / 16 / 8, 256, 0, stream>>>(feats, ws, out);
}